// Point_Transformer_79070347919650
// MI455X (gfx1250) — compile-verified
//
#include <hip/hip_runtime.h>
#include <hip/hip_bf16.h>

typedef __attribute__((ext_vector_type(2))) float v2f;
typedef __attribute__((ext_vector_type(8))) float v8f;

#define CCH   128
#define NPTS  4096
#define BATCH 4
#define KNN   32
#define ROWS  (BATCH * NPTS)          // 16384 rows for anchor-side and neighbor-side
#define R2    (0.12f * 0.12f)

// workspace layout (in floats)
#define WS_WQA 0L                      // folded Wq@Wa, padded to 4x128 (row 3 = 0)
#define WS_WKA 512L                    // folded Wk@Wa, padded 4x128
#define WS_WDP 1024L                   // Wd, padded 4x128
#define WS_C0  1536L                   // (bq-bk)@Wa + ba
#define WS_ZB  1664L                   // 128 zeros (bias for Lnb/Dnb)
#define WS_LA  2048L                   // anchor@Wqa + c0        (ROWS x 128)
#define WS_DA  (WS_LA  + (long)ROWS * CCH)   // anchor@Wd + bd
#define WS_LNB (WS_DA  + (long)ROWS * CCH)   // neighbor@Wka
#define WS_DNB (WS_LNB + (long)ROWS * CCH)   // neighbor@Wd

// ---------------------------------------------------------------------------
// Kernel 0: fold weights.  Wqa = Wq@Wa, Wka = Wk@Wa, c0 = (bq-bk)@Wa + ba.
// One block of 128 threads; thread c owns output column c.
// ---------------------------------------------------------------------------
__global__ void pt_prep_kernel(const float* __restrict__ Wq, const float* __restrict__ bq,
                               const float* __restrict__ Wk, const float* __restrict__ bk,
                               const float* __restrict__ Wd,
                               const float* __restrict__ Wa, const float* __restrict__ ba,
                               float* __restrict__ ws) {
  const int c = threadIdx.x;           // 0..127
  float* Wqa = ws + WS_WQA;
  float* Wka = ws + WS_WKA;
  float* Wdp = ws + WS_WDP;
  float* c0  = ws + WS_C0;
  float* zb  = ws + WS_ZB;

  for (int r = 0; r < 3; ++r) {
    float sq = 0.f, sk = 0.f;
    for (int j = 0; j < CCH; ++j) {
      const float wa = Wa[j * CCH + c];
      sq += Wq[r * CCH + j] * wa;
      sk += Wk[r * CCH + j] * wa;
    }
    Wqa[r * CCH + c] = sq;
    Wka[r * CCH + c] = sk;
    Wdp[r * CCH + c] = Wd[r * CCH + c];
  }
  Wqa[3 * CCH + c] = 0.f;              // K=3 padding row
  Wka[3 * CCH + c] = 0.f;
  Wdp[3 * CCH + c] = 0.f;
  zb[c] = 0.f;

  float s0 = 0.f;
  for (int j = 0; j < CCH; ++j) s0 += (bq[j] - bk[j]) * Wa[j * CCH + c];
  c0[c] = s0 + ba[c];
}

// ---------------------------------------------------------------------------
// Kernel 1: projections via V_WMMA_F32_16X16X4_F32.
//   D(16x16) = A(16 points x [x,y,z,0]) * B(4x16 weight tile) + bias
// grid = (128 row-groups, 4 arrays); block = 256 threads = 8 waves.
// Wave w owns column tile w and keeps the B operand + bias resident while
// looping over 8 row tiles (8 WMMAs per wave, fully unrolled).
// A layout (ISA 7.12.2, 32-bit A 16x4): lanes 0-15 hold K=0(v0)/K=1(v1),
// lanes 16-31 hold K=2(v0)/K=3(v1).  B/C/D: one row striped across lanes.
// All loads are unconditional (value-selects, no exec-mask branches).
// ---------------------------------------------------------------------------
__global__ void pt_proj_kernel(const float* __restrict__ anchor,
                               const float* __restrict__ nbr,
                               const float* __restrict__ bd,
                               float* __restrict__ ws) {
  const int wave    = threadIdx.x >> 5;
  const int lane    = threadIdx.x & 31;
  const int r       = lane & 15;
  const bool hi     = lane >= 16;
  const int col0    = wave * 16;
  const int rowbase = blockIdx.x * 128;

  const float* in; const float* W; const float* bias; float* outp;
  switch (blockIdx.y) {
    case 0:  in = anchor; W = ws + WS_WQA; bias = ws + WS_C0; outp = ws + WS_LA;  break;
    case 1:  in = anchor; W = ws + WS_WDP; bias = bd;         outp = ws + WS_DA;  break;
    case 2:  in = nbr;    W = ws + WS_WKA; bias = ws + WS_ZB; outp = ws + WS_LNB; break;
    default: in = nbr;    W = ws + WS_WDP; bias = ws + WS_ZB; outp = ws + WS_DNB; break;
  }

  // B operand (resident across all 8 row tiles): rows K=0/1 for lanes 0-15,
  // K=2/3 for lanes 16-31; the K=3 row is zero.
  const float b0  = W[(hi ? 2 : 0) * CCH + col0 + r];
  const float b1t = W[1 * CCH + col0 + r];
  v2f Bv;
  Bv[0] = b0;
  Bv[1] = hi ? 0.f : b1t;
  const float bv = bias[col0 + r];

  #pragma unroll
  for (int t = 0; t < 8; ++t) {
    const int row0 = rowbase + t * 16;
    const float* p = in + (long)(row0 + r) * 3;
    const float p0 = p[0], p1 = p[1], p2 = p[2];
    v2f A;
    A[0] = hi ? p2  : p0;
    A[1] = hi ? 0.f : p1;

    v8f acc = {};
    acc = __builtin_amdgcn_wmma_f32_16x16x4_f32(
        /*neg_a=*/false, A, /*neg_b=*/false, Bv,
        /*c_mod=*/(short)0, acc, /*reuse_a=*/false, /*reuse_b=*/false);

    // D element (m = v + 8*hi, n = r)
    #pragma unroll
    for (int v = 0; v < 8; ++v) {
      const int m = v + (hi ? 8 : 0);
      outp[(long)(row0 + m) * CCH + col0 + r] = acc[v] + bv;
    }
  }
}

// ---------------------------------------------------------------------------
// Kernel 2: fused ball query + softmax(C) + max(K).
// One block = 256 threads = 8 waves, handles one batch slice of 64 anchors.
// Neighbor xyz for the batch lives in LDS (48 KB). One wave per anchor:
//   - scan 32 neighbors/iter, ballot_w32 + popcount-prefix packs hit indices
//     in ascending-m order, early exit at 32 hits (matches pointnet2 pad rule)
//   - channels: 4 per lane (c = lane + 32*j); softmax over C via shfl_xor
//     butterflies; running max over K.
// ---------------------------------------------------------------------------
__global__ void pt_attn_kernel(const float* __restrict__ anchor,
                               const float* __restrict__ nbr,
                               const float* __restrict__ ws,
                               float* __restrict__ out) {
  __shared__ float snb[NPTS * 3];
  __shared__ int   sidx[8][KNN];

  const int blocksPerBatch = NPTS / 64;            // 64
  const int b     = blockIdx.x / blocksPerBatch;
  const int chunk = blockIdx.x % blocksPerBatch;
  const int wave  = threadIdx.x >> 5;
  const int lane  = threadIdx.x & 31;

  const float* nbB = nbr + (long)b * NPTS * 3;
  for (int i = threadIdx.x; i < NPTS * 3; i += blockDim.x) snb[i] = nbB[i];
  __syncthreads();

  const float* La  = ws + WS_LA;
  const float* Da  = ws + WS_DA;
  const float* Lnb = ws + WS_LNB;
  const float* Dnb = ws + WS_DNB;

  for (int a = 0; a < 8; ++a) {
    const int n  = chunk * 64 + wave * 8 + a;       // anchor within batch
    const long gn = (long)b * NPTS + n;             // global row
    const float ax = anchor[gn * 3 + 0];
    const float ay = anchor[gn * 3 + 1];
    const float az = anchor[gn * 3 + 2];

    // ---- ball query: first KNN hits in ascending neighbor order ----
    int found = 0;
    for (int m0 = 0; m0 < NPTS && found < KNN; m0 += 32) {
      const int m = m0 + lane;
      const float dx = ax - snb[m * 3 + 0];
      const float dy = ay - snb[m * 3 + 1];
      const float dz = az - snb[m * 3 + 2];
      const bool hit = (dx * dx + dy * dy + dz * dz) < R2;
      const unsigned bal = __builtin_amdgcn_ballot_w32(hit);
      const int slot = found + __popc(bal & ((1u << lane) - 1u));
      if (hit && slot < KNN) sidx[wave][slot] = m;
      found += __popc(bal);
    }
    __syncthreads();
    const int valid = found < KNN ? found : KNN;
    const int pad   = (valid > 0) ? sidx[wave][0] : 0;
    if (lane >= valid) sidx[wave][lane] = pad;      // pad with first hit (or 0)
    __syncthreads();

    // ---- attention: softmax over C, max over K ----
    const float* Lar = La + gn * CCH;
    const float* Dar = Da + gn * CCH;
    float la[4], da[4], accv[4];
    #pragma unroll
    for (int j = 0; j < 4; ++j) {
      la[j] = Lar[lane + 32 * j];
      da[j] = Dar[lane + 32 * j];
      accv[j] = -__builtin_inff();
    }

    for (int k = 0; k < KNN; ++k) {
      const int m = sidx[wave][k];
      const float* Lr = Lnb + ((long)b * NPTS + m) * CCH;
      const float* Dr = Dnb + ((long)b * NPTS + m) * CCH;

      float l[4];
      float mx = -__builtin_inff();
      #pragma unroll
      for (int j = 0; j < 4; ++j) {
        l[j] = la[j] - Lr[lane + 32 * j];
        mx = fmaxf(mx, l[j]);
      }
      #pragma unroll
      for (int off = 16; off > 0; off >>= 1)
        mx = fmaxf(mx, __shfl_xor(mx, off, 32));

      float e[4];
      float s = 0.f;
      #pragma unroll
      for (int j = 0; j < 4; ++j) { e[j] = __expf(l[j] - mx); s += e[j]; }
      #pragma unroll
      for (int off = 16; off > 0; off >>= 1)
        s += __shfl_xor(s, off, 32);
      const float inv = 1.0f / s;

      #pragma unroll
      for (int j = 0; j < 4; ++j) {
        const float d = da[j] - Dr[lane + 32 * j];
        accv[j] = fmaxf(accv[j], d * (e[j] * inv));
      }
    }
    #pragma unroll
    for (int j = 0; j < 4; ++j) out[gn * CCH + lane + 32 * j] = accv[j];
    __syncthreads();
  }
}

extern "C" void kernel_launch(void* const* d_in, const int* in_sizes, int n_in,
                              void* d_out, int out_size, void* d_ws, size_t ws_size,
                              hipStream_t stream) {
  const float* anchor = (const float*)d_in[0];   // (4,4096,3)
  const float* nbr    = (const float*)d_in[1];   // (4,4096,3)
  const float* Wq     = (const float*)d_in[2];
  const float* bq     = (const float*)d_in[3];
  const float* Wk     = (const float*)d_in[4];
  const float* bk     = (const float*)d_in[5];
  const float* Wd     = (const float*)d_in[6];
  const float* bd     = (const float*)d_in[7];
  const float* Wa     = (const float*)d_in[8];
  const float* ba     = (const float*)d_in[9];
  float* ws  = (float*)d_ws;                     // needs ~33.6 MB
  float* out = (float*)d_out;                    // (4,4096,128) f32

  pt_prep_kernel<<<1, 128, 0, stream>>>(Wq, bq, Wk, bk, Wd, Wa, ba, ws);
  pt_proj_kernel<<<dim3(ROWS / 128, 4), 256, 0, stream>>>(anchor, nbr, bd, ws);
  pt_attn_kernel<<<BATCH * (NPTS / 64), 256, 0, stream>>>(anchor, nbr, ws, out);
}